// MultiHeadAttention_5729486373348
// MI455X (gfx1250) — compile-verified
//
#include <hip/hip_runtime.h>
#include <hip/hip_bf16.h>

#define BATCH   8
#define SEQ     1024
#define DMODEL  512
#define NHEAD   8
#define HDIM    64

typedef __attribute__((ext_vector_type(2))) float v2f;
typedef __attribute__((ext_vector_type(8))) float v8f;

__device__ __forceinline__ v8f wmma_f32(v2f a, v2f b, v8f c) {
    // D = A(16x4 f32) * B(4x16 f32) + C(16x16 f32)
    return __builtin_amdgcn_wmma_f32_16x16x4_f32(
        /*neg_a=*/false, a, /*neg_b=*/false, b,
        /*c_mod=*/(short)0, c, /*reuse_a=*/false, /*reuse_b=*/false);
}

// ---------------------------------------------------------------------------
// Kernel 1: per-head QKV projections.
//   [B*S, D] x [D, H*HD] (+bias) -> Qp/Kp/Vp stored [B,H,S,HD]
// One wave per 16x16 output tile; 128 WMMA_F32_16X16X4 steps over D=512.
// ---------------------------------------------------------------------------
__global__ __launch_bounds__(256) void qkv_proj_kernel(
    const float* __restrict__ q, const float* __restrict__ k,
    const float* __restrict__ v,
    const float* __restrict__ Wq, const float* __restrict__ bq,
    const float* __restrict__ Wk, const float* __restrict__ bk,
    const float* __restrict__ Wv, const float* __restrict__ bv,
    float* __restrict__ Qp, float* __restrict__ Kp, float* __restrict__ Vp)
{
    const int wid  = threadIdx.x >> 5;
    const int lane = threadIdx.x & 31;
    const int n    = lane & 15;   // row of A / col of B,C,D
    const int hh   = lane >> 4;   // half-wave select

    int tile   = blockIdx.x * 8 + wid;        // 0 .. 3*16384-1
    int tensor = tile / 16384;
    int t      = tile - tensor * 16384;
    int rbase  = (t >> 5) << 4;               // row in flattened B*S
    int cglob  = (t & 31) << 4;               // col in H*HD
    int h      = cglob >> 6;
    int cbase  = cglob & (HDIM - 1);

    const float *X, *W, *bias;
    float* Out;
    if (tensor == 0)      { X = q; W = Wq; bias = bq; Out = Qp; }
    else if (tensor == 1) { X = k; W = Wk; bias = bk; Out = Kp; }
    else                  { X = v; W = Wv; bias = bv; Out = Vp; }

    const float* xrow = X + (size_t)(rbase + n) * DMODEL + 2 * hh;
    const float* wcol = W + ((size_t)h * DMODEL + 2 * hh) * HDIM + cbase + n;

    v8f acc = {};
    #pragma unroll 4
    for (int kk0 = 0; kk0 < DMODEL; kk0 += 4) {
        v2f a = *(const v2f*)(xrow + kk0);
        v2f bb;
        bb[0] = wcol[(size_t)kk0 * HDIM];
        bb[1] = wcol[(size_t)(kk0 + 1) * HDIM];
        acc = wmma_f32(a, bb, acc);
    }

    const float bval = bias[h * HDIM + cbase + n];
    #pragma unroll
    for (int r = 0; r < 8; ++r) {
        int rflat = rbase + r + 8 * hh;
        int bidx  = rflat >> 10;          // / SEQ
        int s     = rflat & (SEQ - 1);
        Out[(((size_t)(bidx * NHEAD + h) * SEQ + s) * HDIM) + cbase + n] =
            acc[r] + bval;
    }
}

// ---------------------------------------------------------------------------
// Kernel 2: fully fused attention for one (b, h, 16-query-row stripe):
//   scores (WMMA) -> mask -> softmax -> attn store (single pass over the
//   268MB attn tensor) -> attn@V (WMMA, prob fragments transposed via
//   wave-private LDS staging) -> ds_add_f32 cross-wave combine -> ctx store.
// 8 waves / block; wave w owns keys [128w, 128w+128).
// ---------------------------------------------------------------------------
__global__ __launch_bounds__(256) void attn_fused_kernel(
    const float* __restrict__ Qp, const float* __restrict__ Kp,
    const float* __restrict__ Vp, const int* __restrict__ mask,
    float* __restrict__ attn, float* __restrict__ ctx)
{
    __shared__ float ldsQ[16 * 66];        // Q tile, padded stride
    __shared__ float ldsP[8][16 * 66];     // per-wave prob staging (16x64)
    __shared__ float ctxAcc[16 * 64];      // cross-wave ctx accumulator
    __shared__ float red_max[8][16];
    __shared__ float red_sum[8][16];

    const int tid  = threadIdx.x;
    const int wid  = tid >> 5;
    const int lane = tid & 31;
    const int n    = lane & 15;
    const int hh   = lane >> 4;

    const int stile = blockIdx.x & 63;
    const int h     = (blockIdx.x >> 6) & 7;
    const int b     = blockIdx.x >> 9;
    const size_t bhS   = (size_t)(b * NHEAD + h) * SEQ;
    const int    srow0 = stile << 4;

    // zero ctx accumulator + stage Q tile [16][64] into LDS
    *(float4*)&ctxAcc[tid << 2] = make_float4(0.f, 0.f, 0.f, 0.f);
    {
        int r  = tid >> 4;
        int c4 = (tid & 15) << 2;
        const float4 qv = *(const float4*)(Qp + (bhS + srow0 + r) * HDIM + c4);
        ldsQ[r * 66 + c4 + 0] = qv.x;
        ldsQ[r * 66 + c4 + 1] = qv.y;
        ldsQ[r * 66 + c4 + 2] = qv.z;
        ldsQ[r * 66 + c4 + 3] = qv.w;
    }
    __syncthreads();

    const int tbase = wid << 7;                    // 128 keys per wave
    const float* qlds = ldsQ + n * 66 + 2 * hh;    // A-frag row = lane&15
    v8f accs[8];

    // ---- scores: Q (LDS) x K^T (global), 128 WMMAs per wave ----
    #pragma unroll
    for (int kt = 0; kt < 8; ++kt) {
        v8f acc = {};
        const int t0 = tbase + (kt << 4);
        const float* krow = Kp + (bhS + t0 + n) * HDIM + 2 * hh;
        #pragma unroll 4
        for (int step = 0; step < 16; ++step) {
            int kk0 = step << 2;
            v2f a  = *(const v2f*)(qlds + kk0);
            v2f bb = *(const v2f*)(krow + kk0);
            acc = wmma_f32(a, bb, acc);
        }
        accs[kt] = acc;
    }

    // ---- scale then mask (masked positions become exactly -10000) ----
    const float scale = 0.125f;  // 1/sqrt(64)
    #pragma unroll
    for (int kt = 0; kt < 8; ++kt) {
        int t0 = tbase + (kt << 4);
        #pragma unroll
        for (int r = 0; r < 8; ++r) {
            int srow = srow0 + r + 8 * hh;
            int mv = mask[((size_t)b * SEQ + srow) * SEQ + t0 + n];
            float sc = accs[kt][r] * scale;
            accs[kt][r] = (mv == 0) ? -10000.0f : sc;
        }
    }

    // ---- row max: shfl within 16-lane half, LDS combine across waves ----
    float mx[8];
    #pragma unroll
    for (int r = 0; r < 8; ++r) {
        float m = accs[0][r];
        #pragma unroll
        for (int kt = 1; kt < 8; ++kt) m = fmaxf(m, accs[kt][r]);
        m = fmaxf(m, __shfl_xor(m, 1));
        m = fmaxf(m, __shfl_xor(m, 2));
        m = fmaxf(m, __shfl_xor(m, 4));
        m = fmaxf(m, __shfl_xor(m, 8));
        mx[r] = m;
    }
    if (n == 0) {
        #pragma unroll
        for (int r = 0; r < 8; ++r) red_max[wid][r + 8 * hh] = mx[r];
    }
    __syncthreads();
    float gmax[8];
    #pragma unroll
    for (int r = 0; r < 8; ++r) {
        float m = red_max[0][r + 8 * hh];
        #pragma unroll
        for (int w2 = 1; w2 < 8; ++w2) m = fmaxf(m, red_max[w2][r + 8 * hh]);
        gmax[r] = m;
    }

    // ---- exp + row sum ----
    float psum[8];
    #pragma unroll
    for (int r = 0; r < 8; ++r) {
        float s = 0.0f;
        #pragma unroll
        for (int kt = 0; kt < 8; ++kt) {
            float e = __expf(accs[kt][r] - gmax[r]);
            accs[kt][r] = e;
            s += e;
        }
        s += __shfl_xor(s, 1);
        s += __shfl_xor(s, 2);
        s += __shfl_xor(s, 4);
        s += __shfl_xor(s, 8);
        psum[r] = s;
    }
    if (n == 0) {
        #pragma unroll
        for (int r = 0; r < 8; ++r) red_sum[wid][r + 8 * hh] = psum[r];
    }
    __syncthreads();

    // ---- normalize in-register + single-pass attn store ----
    #pragma unroll
    for (int r = 0; r < 8; ++r) {
        float s = red_sum[0][r + 8 * hh];
        #pragma unroll
        for (int w2 = 1; w2 < 8; ++w2) s += red_sum[w2][r + 8 * hh];
        const float inv = 1.0f / s;
        int srow = srow0 + r + 8 * hh;
        float* arow = attn + (bhS + srow) * SEQ;
        #pragma unroll
        for (int kt = 0; kt < 8; ++kt) {
            float p = accs[kt][r] * inv;
            accs[kt][r] = p;
            arow[tbase + (kt << 4) + n] = p;
        }
    }

    // ---- attn@V: stage probs (wave-private LDS) as two 16x64 halves and
    //      accumulate this wave's 16x64 partial with 128 more WMMAs ----
    v8f o0 = {}, o1 = {}, o2 = {}, o3 = {};
    float* myP = ldsP[wid];
    #pragma unroll
    for (int hc = 0; hc < 2; ++hc) {
        #pragma unroll
        for (int kt4 = 0; kt4 < 4; ++kt4) {
            int kt = hc * 4 + kt4;
            #pragma unroll
            for (int r = 0; r < 8; ++r)
                myP[(r + 8 * hh) * 66 + (kt4 << 4) + n] = accs[kt][r];
        }
        // wave-private staging: DS ops within a wave stay in order
        const float* prow = myP + n * 66 + 2 * hh;
        const float* vb = Vp + (bhS + tbase + hc * 64 + 2 * hh) * HDIM + n;
        #pragma unroll 4
        for (int step = 0; step < 16; ++step) {
            int t0 = step << 2;
            v2f a = *(const v2f*)(prow + t0);
            const float* vp0 = vb + (size_t)t0 * HDIM;
            v2f b0, b1, b2, b3;
            b0[0] = vp0[0];       b0[1] = vp0[HDIM + 0];
            b1[0] = vp0[16];      b1[1] = vp0[HDIM + 16];
            b2[0] = vp0[32];      b2[1] = vp0[HDIM + 32];
            b3[0] = vp0[48];      b3[1] = vp0[HDIM + 48];
            o0 = wmma_f32(a, b0, o0);
            o1 = wmma_f32(a, b1, o1);
            o2 = wmma_f32(a, b2, o2);
            o3 = wmma_f32(a, b3, o3);
        }
    }

    // ---- cross-wave combine via ds_add_f32 ----
    #pragma unroll
    for (int r = 0; r < 8; ++r) {
        int row = (r + 8 * hh) << 6;   // *64
        atomicAdd(&ctxAcc[row + n],      o0[r]);
        atomicAdd(&ctxAcc[row + 16 + n], o1[r]);
        atomicAdd(&ctxAcc[row + 32 + n], o2[r]);
        atomicAdd(&ctxAcc[row + 48 + n], o3[r]);
    }
    __syncthreads();

    // ---- ctx store, head-concat layout [B,S,512] ----
    {
        int r  = tid >> 4;
        int c4 = (tid & 15) << 2;
        float4 val = *(float4*)&ctxAcc[(r << 6) + c4];
        *(float4*)(ctx + ((size_t)b * SEQ + srow0 + r) * DMODEL + h * HDIM + c4) = val;
    }
}

// ---------------------------------------------------------------------------
// Kernel 3: proj = ctx @ Wo + bo.  [8192,512] x [512,512] tiled GEMM.
// ---------------------------------------------------------------------------
__global__ __launch_bounds__(256) void out_proj_kernel(
    const float* __restrict__ ctx, const float* __restrict__ Wo,
    const float* __restrict__ bo, float* __restrict__ proj)
{
    const int wid  = threadIdx.x >> 5;
    const int lane = threadIdx.x & 31;
    const int n    = lane & 15;
    const int hh   = lane >> 4;

    int tile  = blockIdx.x * 8 + wid;    // 0..16383
    int rbase = (tile >> 5) << 4;
    int cbase = (tile & 31) << 4;

    const float* xrow = ctx + (size_t)(rbase + n) * DMODEL + 2 * hh;
    const float* wcol = Wo + (size_t)(2 * hh) * DMODEL + cbase + n;

    v8f acc = {};
    #pragma unroll 4
    for (int kk0 = 0; kk0 < DMODEL; kk0 += 4) {
        v2f a = *(const v2f*)(xrow + kk0);
        v2f bb;
        bb[0] = wcol[(size_t)kk0 * DMODEL];
        bb[1] = wcol[(size_t)(kk0 + 1) * DMODEL];
        acc = wmma_f32(a, bb, acc);
    }

    const float bias = bo[cbase + n];
    #pragma unroll
    for (int r = 0; r < 8; ++r)
        proj[(size_t)(rbase + r + 8 * hh) * DMODEL + cbase + n] = acc[r] + bias;
}

// ---------------------------------------------------------------------------
extern "C" void kernel_launch(void* const* d_in, const int* in_sizes, int n_in,
                              void* d_out, int out_size, void* d_ws, size_t ws_size,
                              hipStream_t stream)
{
    const float* q    = (const float*)d_in[0];
    const float* k    = (const float*)d_in[1];
    const float* v    = (const float*)d_in[2];
    const int*   mask = (const int*)  d_in[3];
    const float* Wq   = (const float*)d_in[4];
    const float* bq   = (const float*)d_in[5];
    const float* Wk   = (const float*)d_in[6];
    const float* bk   = (const float*)d_in[7];
    const float* Wv   = (const float*)d_in[8];
    const float* bv   = (const float*)d_in[9];
    const float* Wo   = (const float*)d_in[10];
    const float* bo   = (const float*)d_in[11];

    float* proj = (float*)d_out;                                  // [B,S,D]
    float* attn = proj + (size_t)BATCH * SEQ * DMODEL;            // [B,H,S,S]

    float* ws  = (float*)d_ws;
    const size_t QKV = (size_t)BATCH * NHEAD * SEQ * HDIM;        // 4,194,304
    float* Qp  = ws;
    float* Kp  = ws + QKV;
    float* Vp  = ws + 2 * QKV;
    float* ctx = ws + 3 * QKV;                                    // [B,S,D]

    // 3 tensors * 16384 tiles / 8 waves-per-block
    qkv_proj_kernel<<<6144, 256, 0, stream>>>(q, k, v, Wq, bq, Wk, bk, Wv, bv,
                                              Qp, Kp, Vp);
    // one block per (b,h,stile): 8*8*64  — fused scores/softmax/attn@V
    attn_fused_kernel<<<4096, 256, 0, stream>>>(Qp, Kp, Vp, mask, attn, ctx);
    // 16384 tiles / 8
    out_proj_kernel<<<2048, 256, 0, stream>>>(ctx, Wo, bo, proj);
}